// VelvetNoise_46729244180795
// MI455X (gfx1250) — compile-verified
//
#include <hip/hip_runtime.h>
#include <stdint.h>

// Problem constants (from reference): B=32, PS=4000, D=16, N=96000, step=24.
constexpr int kB    = 32;
constexpr int kPS   = 4000;
constexpr int kD    = 16;
constexpr int kStep = 24;                 // N / PS
constexpr int kWinPerWave  = 2;           // windows (pulses) per wave32
constexpr int kWavesPerBlk = 8;
constexpr int kWinPerBlock = kWinPerWave * kWavesPerBlk;  // 16
constexpr int kThreads     = 32 * kWavesPerBlk;           // 256
constexpr int kWaveFloats  = kWinPerWave * kStep * kD;    // 2*24*16 = 768 floats = 3KB
constexpr int kTileFloats  = kWavesPerBlk * kWaveFloats;  // 6144 floats = 24KB
constexpr int kIters       = (kWaveFloats / 4) / 32;      // 6 float4 per lane

__global__ __launch_bounds__(kThreads)
void velvet_scatter_kernel(const float* __restrict__ in, float* __restrict__ out) {
    // 24KB LDS tile, split into 8 independent per-wave 3KB regions.
    __shared__ __align__(16) float tile[kTileFloats];

    const int tid  = threadIdx.x;
    const int wave = tid >> 5;      // 0..7
    const int lane = tid & 31;

    // First window owned by this wave. PS % 16 == 0 -> a block never crosses a
    // batch row, so each wave's input (256B) and output (3KB) are contiguous.
    const long long w0 =
        (long long)blockIdx.x * kWinPerBlock + (long long)wave * kWinPerWave;

    float* wtile = tile + wave * kWaveFloats;

    // ---- 1) zero this wave's region (DS ops are in-order within a wave) ----
    float4* t4 = (float4*)wtile;
#pragma unroll
    for (int k = 0; k < kIters; ++k) {
        t4[lane + 32 * k] = float4{0.f, 0.f, 0.f, 0.f};
    }

    // ---- 2) scatter: one (window, channel) pair per lane ----
    const int wi = lane >> 4;       // window within wave, 0..1
    const int d  = lane & 15;       // channel, 0..15
    const float* inb = in + w0 * (2 * kD);
    const float sgn  = inb[wi * (2 * kD) + d];
    const float frac = inb[wi * (2 * kD) + kD + d];
    // jitter = trunc((step-1) * frac) in [0, 22]  (matches astype(int32))
    const int j = (int)((float)(kStep - 1) * frac);
    wtile[wi * (kStep * kD) + j * kD + d] = sgn;

    // Wave-local: ensure DS stores committed before async engine reads LDS.
    asm volatile("s_wait_dscnt 0x0" ::: "memory");

    // ---- 3) stream region -> global via async LDS->memory b128 stores ----
    const uint64_t gbase    = (uint64_t)(uintptr_t)(out + w0 * (kStep * kD));
    const uint32_t lds_base = (uint32_t)(uintptr_t)wtile;  // low 32 bits = LDS offset
#pragma unroll
    for (int k = 0; k < kIters; ++k) {
        const uint32_t byte_off = (uint32_t)(lane + 32 * k) * 16u;
        const uint32_t lds_addr = lds_base + byte_off;
        // GVS mode: mem_addr = SADDR(64) + VADDR(32) ; VSRC holds LDS byte addr.
        asm volatile("global_store_async_from_lds_b128 %0, %1, %2"
                     :
                     : "v"(byte_off), "v"(lds_addr), "s"(gbase)
                     : "memory");
    }
    // Drain ASYNCcnt before wave exit.
    asm volatile("s_wait_asynccnt 0x0" ::: "memory");
}

extern "C" void kernel_launch(void* const* d_in, const int* in_sizes, int n_in,
                              void* d_out, int out_size, void* d_ws, size_t ws_size,
                              hipStream_t stream) {
    (void)in_sizes; (void)n_in; (void)d_ws; (void)ws_size; (void)out_size;
    const float* in = (const float*)d_in[0];   // (B, PS, 2D) float32
    float* out      = (float*)d_out;           // (B, N, D)  float32

    const int total_windows = kB * kPS;                    // 128000
    const int blocks = total_windows / kWinPerBlock;       // 8000
    velvet_scatter_kernel<<<blocks, kThreads, 0, stream>>>(in, out);
}